// SymbolicTraversal_73641509257592
// MI455X (gfx1250) — compile-verified
//
#include <hip/hip_runtime.h>

// -------------------------------------------------------------------------
// SymbolicTraversal: tropical (max,*) pattern-matched sparse traversal.
// CDNA5 path: double-buffered GLOBAL_LOAD_ASYNC_TO_LDS_B128 staging of the
// 8MB (rel,ts) key stream (ASYNCcnt pipeline), LDS bitmap filter, int
// global_atomic_max scatter. WMMA is inapplicable (max-semiring, sparse).
// -------------------------------------------------------------------------

#define MAX_B           32
#define BITMAP_WORDS    128   // key = (rel<<5)|ts, rel<100, ts<20 -> key>>5 < 100
#define TILE            256   // edges per buffer per wave (1KB per array)
#define WAVES_PER_BLOCK 8
#define THREADS         (WAVES_PER_BLOCK * 32)
#define WAVE_LDS_INTS   (2 * 2 * TILE)   // 2 buffers x {rel,ts} x TILE

__global__ __launch_bounds__(256)
void st_init_out(float4* __restrict__ out4, int n4) {
    int i = blockIdx.x * blockDim.x + threadIdx.x;
    if (i < n4) out4[i] = make_float4(0.f, 0.f, 0.f, 0.f);
}

// One array chunk: 1KB = 2 x b128 per lane (32 lanes x 16B x 2).
// INST_OFFSET is added to BOTH the LDS and global addresses (ISA 08 §4.4).
__device__ __forceinline__
void async_fill_1kb(unsigned ldsaddr, unsigned goff, const int* base) {
    asm volatile(
        "global_load_async_to_lds_b128 %0, %1, %2 offset:0\n\t"
        "global_load_async_to_lds_b128 %0, %1, %2 offset:512"
        :: "v"(ldsaddr), "v"(goff), "s"(base)
        : "memory");
}

__device__ __forceinline__ void wait_async_4()  {
    asm volatile("s_wait_asynccnt 0x4" ::: "memory");
}
__device__ __forceinline__ void wait_async_0()  {
    asm volatile("s_wait_asynccnt 0x0" ::: "memory");
}

__global__ __launch_bounds__(THREADS)
void st_traverse(const float* __restrict__ h_prob,   // [B, N]
                 const float* __restrict__ ew,       // [E]
                 const int*   __restrict__ eh,       // [E]
                 const int*   __restrict__ et,       // [E]
                 const int*   __restrict__ er,       // [E]
                 const int*   __restrict__ ets,      // [E]
                 const int*   __restrict__ r_index,  // [B]
                 const int*   __restrict__ tstamp,   // [B]
                 float*       __restrict__ out,      // [B, N]
                 int E, int N, int B) {
    __shared__ int      s_key[MAX_B];
    __shared__ unsigned s_bits[BITMAP_WORDS];
    __shared__ int      s_stage[WAVES_PER_BLOCK * WAVE_LDS_INTS];

    const int t    = threadIdx.x;
    const int lane = t & 31;          // wave32
    const int wid  = t >> 5;

    // ---- Build active-pattern bitmap + exact key table (once per block) ----
    if (t < BITMAP_WORDS) s_bits[t] = 0u;
    __syncthreads();
    if (t < B) {
        int k = (r_index[t] << 5) | tstamp[t];
        s_key[t] = k;
        atomicOr(&s_bits[(k >> 5) & (BITMAP_WORDS - 1)], 1u << (k & 31));
    }
    __syncthreads();

    int* __restrict__ iout = reinterpret_cast<int*>(out);

    // Hit path: exact-key match -> gather h_prob -> int atomic max (v >= 0).
    auto process = [&](int k, int e) {
        if (s_bits[(k >> 5) & (BITMAP_WORDS - 1)] & (1u << (k & 31))) {
            const int   head = eh[e];
            const int   tail = et[e];
            const float w    = ew[e];
            for (int b = 0; b < B; ++b) {
                if (s_key[b] == k) {
                    float v = h_prob[b * N + head] * w;
                    atomicMax(&iout[b * N + tail], __float_as_int(v));
                }
            }
        }
    };

    // ---- Wave-private double-buffered async staging of (rel, ts) keys ----
    int* stage = &s_stage[wid * WAVE_LDS_INTS];
    // Low 32 bits of the generic pointer == wave-relative LDS byte offset.
    const unsigned lds_base  = (unsigned)(size_t)(void*)stage;
    const unsigned lane16    = (unsigned)(lane * 16);

    const int fullTiles = E / TILE;
    const int gwave     = blockIdx.x * WAVES_PER_BLOCK + wid;
    const int nwaves    = gridDim.x * WAVES_PER_BLOCK;

    auto fill = [&](int buf, int tile) {
        unsigned goff = (unsigned)(tile * TILE) * 4u + lane16;
        unsigned ldsA = lds_base + (unsigned)(buf * (2 * TILE * 4)) + lane16;
        async_fill_1kb(ldsA,                      goff, er );  // rel chunk
        async_fill_1kb(ldsA + (unsigned)(TILE*4), goff, ets);  // ts  chunk
    };

    if (gwave < fullTiles) {
        fill(0, gwave);                       // prefill buffer 0
        int parity = 0;
        for (int tl = gwave; tl < fullTiles; tl += nwaves) {
            const int nx = tl + nwaves;
            if (nx < fullTiles) {
                fill(parity ^ 1, nx);         // overlap next tile's DMA
                wait_async_4();               // in-order: current 4 ops done
            } else {
                wait_async_0();
            }
            // Consume current buffer: stride-32 => bank-conflict-free ds reads.
            const int* rl = stage + parity * (2 * TILE);
            const int* tb = rl + TILE;
            const int  ebase = tl * TILE;
            #pragma unroll 4
            for (int j = lane; j < TILE; j += 32) {
                int k = (rl[j] << 5) | tb[j];
                process(k, ebase + j);
            }
            parity ^= 1;
        }
    }

    // ---- Ragged tail (E % TILE edges): direct loads by global wave 0 ----
    if (gwave == 0) {
        for (int e = fullTiles * TILE + lane; e < E; e += 32) {
            int k = (er[e] << 5) | ets[e];
            process(k, e);
        }
    }
}

extern "C" void kernel_launch(void* const* d_in, const int* in_sizes, int n_in,
                              void* d_out, int out_size, void* d_ws, size_t ws_size,
                              hipStream_t stream) {
    const float* h_prob  = (const float*)d_in[0];
    const float* ew      = (const float*)d_in[1];
    const int*   eh      = (const int*)  d_in[2];
    const int*   et      = (const int*)  d_in[3];
    const int*   er      = (const int*)  d_in[4];
    const int*   ets     = (const int*)  d_in[5];
    const int*   r_index = (const int*)  d_in[6];
    const int*   tstamp  = (const int*)  d_in[7];
    // d_in[8] is num_nodes in DEVICE memory; derive N on host instead.

    const int B = in_sizes[6];               // 32
    const int N = in_sizes[0] / B;           // 20000
    const int E = in_sizes[1];               // 1,000,000

    float* out = (float*)d_out;

    // 1) Zero-init output (int 0 == float 0.0f == atomicMax identity here).
    {
        int n4 = out_size >> 2;              // B*N divisible by 4
        int blocks = (n4 + 255) / 256;
        st_init_out<<<blocks, 256, 0, stream>>>((float4*)out, n4);
    }

    // 2) Async-staged key stream, bitmap filter, scatter-max.
    {
        int blocks = 128;   // 1024 waves -> ~3.8 tiles/wave, pipeline active
        st_traverse<<<blocks, THREADS, 0, stream>>>(h_prob, ew, eh, et, er, ets,
                                                    r_index, tstamp, out, E, N, B);
    }
}